// YOLOLoss_57423712748213
// MI455X (gfx1250) — compile-verified
//
#include <hip/hip_runtime.h>
#include <hip/hip_bf16.h>
#include <math.h>

typedef __attribute__((ext_vector_type(2))) float v2f;
typedef __attribute__((ext_vector_type(8))) float v8f;

#define BSZ   32
#define NANC  3
#define NCLS  80
#define HH    64
#define WWD   64
#define NTGT  50
#define NCELL (BSZ*NANC*HH*WWD)      /* 393216 */
#define NWORD (NCELL/32)             /* 12288  */
#define EPSF  1e-12f

struct WS {
  unsigned int maskbits[NWORD];   // claim map for scattered "mask" cells
  unsigned int noobjbits[NWORD];  // claim map for noobj-zeroed cells
  float acc[16];
  // 0:bce_x 1:bce_y 2:sq_w 3:sq_h 4:conf_pos 5:conf_all 6:conf_corr 7:cls 8:nmask 9:nzero
};

__device__ __forceinline__ float sigm(float x)  { return 1.0f / (1.0f + expf(-x)); }
__device__ __forceinline__ float clip01(float p){ return fminf(fmaxf(p, EPSF), 1.0f - EPSF); }

__global__ void k_init(WS* ws) {
  int i = blockIdx.x * blockDim.x + threadIdx.x;
  if (i < NWORD) { ws->maskbits[i] = 0u; ws->noobjbits[i] = 0u; }
  if (i < 16) ws->acc[i] = 0.0f;
}

// One thread per (batch, target): anchor match, pred-IoU ignore set, and all
// sparse (masked) loss contributions via atomics. Claim bitmaps reproduce
// JAX scatter-set "single writer per duplicate index" semantics.
__global__ void k_targets(const float* __restrict__ inp,
                          const float* __restrict__ tgt, WS* ws) {
  int tid = blockIdx.x * blockDim.x + threadIdx.x;
  if (tid >= BSZ * NTGT) return;
  int b = tid / NTGT, t = tid - b * NTGT;
  const float* g = tgt + (size_t)(b * NTGT + t) * 5;
  float cls = g[0], cx = g[1], cy = g[2], w_ = g[3], h_ = g[4];
  bool valid = (cls + cx + cy + w_ + h_) > 0.0f;
  float gx = cx * WWD, gy = cy * HH, gw = w_ * WWD, gh = h_ * HH;
  int gi = min(max((int)gx, 0), WWD - 1);
  int gj = min(max((int)gy, 0), HH - 1);

  const float aw[3] = {1.25f, 2.0f, 4.125f};    // ANCHORS / (512/64)
  const float ah[3] = {1.625f, 3.75f, 2.875f};

  int best = 0; float bestv = -1.0f;
  for (int n = 0; n < 3; ++n) {                  // anchor-shape IoU argmax (first max wins)
    float inter = fminf(gw, aw[n]) * fminf(gh, ah[n]);
    float iou = inter / (gw * gh + aw[n] * ah[n] - inter + 1e-16f);
    if (iou > bestv) { bestv = iou; best = n; }
  }

  float b1x1 = gx - gw * 0.5f, b1x2 = gx + gw * 0.5f;
  float b1y1 = gy - gh * 0.5f, b1y2 = gy + gh * 0.5f;
  float a1 = (b1x2 - b1x1) * (b1y2 - b1y1);

  // noobj: zero cells where pred-IoU > 0.5 for any anchor; correct conf sum once per cell
  for (int a = 0; a < 3; ++a) {
    size_t base = ((size_t)(b * 255 + a * 85) * HH + gj) * WWD + gi;  // channel stride 4096
    float px = sigm(inp[base + 0 * 4096]);
    float py = sigm(inp[base + 1 * 4096]);
    float pw = inp[base + 2 * 4096];
    float ph = inp[base + 3 * 4096];
    float pbx = px + (float)gi, pby = py + (float)gj;
    float pbw = expf(pw) * aw[a], pbh = expf(ph) * ah[a];
    float b2x1 = pbx - pbw * 0.5f, b2x2 = pbx + pbw * 0.5f;
    float b2y1 = pby - pbh * 0.5f, b2y2 = pby + pbh * 0.5f;
    float iw = fmaxf(fminf(b1x2, b2x2) - fmaxf(b1x1, b2x1), 0.0f);
    float ih = fmaxf(fminf(b1y2, b2y2) - fmaxf(b1y1, b2y1), 0.0f);
    float inter = iw * ih;
    float a2 = (b2x2 - b2x1) * (b2y2 - b2y1);
    float piou = inter / (a1 + a2 - inter + 1e-16f);
    if (valid && piou > 0.5f) {
      int cell = ((b * NANC + a) * HH + gj) * WWD + gi;
      unsigned int bit = 1u << (cell & 31);
      unsigned int old = atomicOr(&ws->noobjbits[cell >> 5], bit);
      if (!(old & bit)) {
        float s = sigm(inp[base + 4 * 4096]);
        atomicAdd(&ws->acc[6], -log1pf(-clip01(s)));
        atomicAdd(&ws->acc[9], 1.0f);
      }
    }
  }

  if (!valid) return;
  int cell = ((b * NANC + best) * HH + gj) * WWD + gi;
  unsigned int bit = 1u << (cell & 31);
  unsigned int old = atomicOr(&ws->maskbits[cell >> 5], bit);
  if (old & bit) return;  // duplicate scatter index: one writer

  size_t base = ((size_t)(b * 255 + best * 85) * HH + gj) * WWD + gi;
  float px = clip01(sigm(inp[base + 0 * 4096]));
  float py = clip01(sigm(inp[base + 1 * 4096]));
  float pw = inp[base + 2 * 4096];
  float ph = inp[base + 3 * 4096];
  float pc = clip01(sigm(inp[base + 4 * 4096]));
  float tx = gx - (float)gi, ty = gy - (float)gj;
  float tw = logf(gw / aw[best] + 1e-16f);
  float th = logf(gh / ah[best] + 1e-16f);
  float gwxh = sigm(gw) * sigm(gh);
  float temp = 2.0f - gwxh; temp *= temp;

  float bx = -(tx * logf(px) + (1.0f - tx) * log1pf(-px));
  float by = -(ty * logf(py) + (1.0f - ty) * log1pf(-py));
  float dw = (pw - tw) * temp, dh = (ph - th) * temp;

  int ci = min(max((int)cls, 0), NCLS - 1);
  float ce = 0.0f;
  for (int c = 0; c < NCLS; ++c) {
    float p = clip01(sigm(inp[base + (size_t)(5 + c) * 4096]));
    ce += (c == ci) ? -logf(p) : -log1pf(-p);
  }

  atomicAdd(&ws->acc[0], bx);
  atomicAdd(&ws->acc[1], by);
  atomicAdd(&ws->acc[2], dw * dw);
  atomicAdd(&ws->acc[3], dh * dh);
  atomicAdd(&ws->acc[4], -logf(pc));
  atomicAdd(&ws->acc[7], ce);
  atomicAdd(&ws->acc[8], 1.0f);
}

// Dense pass over the conf channel only (1.5 MB): Σ -log1p(-clip(sigmoid(conf))).
// In-wave reduction done with V_WMMA_F32_16X16X4_F32: A(16x4) holds the 32 lane
// partials (VGPR0=K{0,2}, VGPR1=0), B=ones => D[m][n] = p_m + p_{m+16}; summing
// the 8 D VGPRs collapses to two half-wave sums, finished by one shuffle.
__global__ void k_conf(const float* __restrict__ inp, WS* ws) {
  const int TOT = 192 * 256;   // total threads; NCELL / TOT = 8 elems each
  int tid = blockIdx.x * blockDim.x + threadIdx.x;
  float psum = 0.0f;
  for (int k = 0; k < 8; ++k) {
    int cell = tid + k * TOT;              // ((b*3+a)<<12) | (j*64+i)
    int ji = cell & 4095;
    int ba = cell >> 12;
    int b = ba / 3, a = ba - b * 3;
    size_t off = ((size_t)(b * 255 + a * 85 + 4) << 12) + ji;
    float s = sigm(inp[off]);
    psum += -log1pf(-clip01(s));
  }
  v2f A; A.x = psum; A.y = 0.0f;
  v2f Bm; Bm.x = 1.0f; Bm.y = 1.0f;
  v8f Cz = {};
  v8f D = __builtin_amdgcn_wmma_f32_16x16x4_f32(
      false, A, false, Bm, (short)0, Cz, false, false);
  float s8 = D[0] + D[1] + D[2] + D[3] + D[4] + D[5] + D[6] + D[7];
  float tot = __shfl(s8, 0) + __shfl(s8, 16);   // low-half sum + high-half sum
  if ((threadIdx.x & 31) == 0) atomicAdd(&ws->acc[5], tot);
}

__global__ void k_final(const WS* __restrict__ ws, float* __restrict__ out) {
  const float N = (float)NCELL;
  const float L0 = -log1pf(-EPSF);              // bce(0,0) baseline per cell
  float nmask = ws->acc[8], nzero = ws->acc[9];
  float lx = (ws->acc[0] + (N - nmask) * L0) / N;
  float ly = (ws->acc[1] + (N - nmask) * L0) / N;
  float lw = ws->acc[2] / N;
  float lh = ws->acc[3] / N;
  float lconf = (ws->acc[4] + (N - nmask) * L0) / N +
                0.5f * ((ws->acc[5] - ws->acc[6] + nzero * L0) / N);
  float lcls = ws->acc[7] / (nmask * (float)NCLS + 1e-16f);
  float loss = 2.5f * (lx + ly) + 2.5f * (lw + lh) + 1.0f * lconf + 1.0f * lcls;
  out[0] = loss; out[1] = lx; out[2] = ly; out[3] = lw;
  out[4] = lh;   out[5] = lconf; out[6] = lcls;
}

extern "C" void kernel_launch(void* const* d_in, const int* in_sizes, int n_in,
                              void* d_out, int out_size, void* d_ws, size_t ws_size,
                              hipStream_t stream) {
  const float* inp = (const float*)d_in[0];   // (32, 255, 64, 64) f32
  const float* tgt = (const float*)d_in[1];   // (32, 50, 5) f32
  float* out = (float*)d_out;                 // 7 scalars
  WS* ws = (WS*)d_ws;                         // ~98 KB used

  k_init   <<<(NWORD + 255) / 256, 256, 0, stream>>>(ws);
  k_targets<<<(BSZ * NTGT + 255) / 256, 256, 0, stream>>>(inp, tgt, ws);
  k_conf   <<<192, 256, 0, stream>>>(inp, ws);
  k_final  <<<1, 1, 0, stream>>>(ws, out);
}